// ImageStatisticsModel_23227183136952
// MI455X (gfx1250) — compile-verified
//
#include <hip/hip_runtime.h>
#include <hip/hip_bf16.h>
#include <math.h>

typedef __attribute__((ext_vector_type(4))) float v4f;
typedef __attribute__((ext_vector_type(2))) float v2f;
typedef __attribute__((ext_vector_type(8))) float v8f;

#define TPB   256
#define HW    (512 * 512)     // spatial elements per (b, c) channel
#define NCH_C 16              // channels per batch

// One workgroup per (b, c) channel. Streams 1 MiB of f32 with non-temporal
// b128 loads, accumulates sum / sumsq / max / min per thread, then reduces:
//   - max/min: wave32 shfl_xor tree + 8-entry LDS
//   - sum/sumsq: V_WMMA_F32_16X16X4_F32 with B = ones folds the 256 thread
//     partials (4 chained WMMAs each), finished by 7 adds + one shfl_xor(16).
__global__ __launch_bounds__(TPB) void image_stats_kernel(
    const float* __restrict__ x, float* __restrict__ out)
{
    const int tid  = threadIdx.x;
    const int lane = tid & 31;
    const int wid  = tid >> 5;

    const float* xb = x + (size_t)blockIdx.x * (size_t)HW;
    const v4f*   xv = (const v4f*)xb;

    // 4 independent accumulator lanes to break dependency chains (ILP)
    float s0 = 0.f, s1 = 0.f, s2 = 0.f, s3 = 0.f;
    float q0 = 0.f, q1 = 0.f, q2 = 0.f, q3 = 0.f;
    float mx0 = -__builtin_inff(), mx1 = mx0, mx2 = mx0, mx3 = mx0;
    float mn0 =  __builtin_inff(), mn1 = mn0, mn2 = mn0, mn3 = mn0;

    // HW/4 = 65536 float4s; 256 iterations per thread, fully coalesced.
#pragma unroll 4
    for (int i = tid; i < HW / 4; i += TPB) {
        v4f v = __builtin_nontemporal_load(xv + i);  // global_load_b128 th:NT
        s0 += v.x;               s1 += v.y;
        s2 += v.z;               s3 += v.w;
        q0 = fmaf(v.x, v.x, q0); q1 = fmaf(v.y, v.y, q1);
        q2 = fmaf(v.z, v.z, q2); q3 = fmaf(v.w, v.w, q3);
        mx0 = fmaxf(mx0, v.x);   mx1 = fmaxf(mx1, v.y);
        mx2 = fmaxf(mx2, v.z);   mx3 = fmaxf(mx3, v.w);
        mn0 = fminf(mn0, v.x);   mn1 = fminf(mn1, v.y);
        mn2 = fminf(mn2, v.z);   mn3 = fminf(mn3, v.w);
    }

    float psum = (s0 + s1) + (s2 + s3);
    float psq  = (q0 + q1) + (q2 + q3);
    float pmax = fmaxf(fmaxf(mx0, mx1), fmaxf(mx2, mx3));
    float pmin = fminf(fminf(mn0, mn1), fminf(mn2, mn3));

    // wave32 shuffle tree for max/min
#pragma unroll
    for (int off = 16; off > 0; off >>= 1) {
        pmax = fmaxf(pmax, __shfl_xor(pmax, off, 32));
        pmin = fminf(pmin, __shfl_xor(pmin, off, 32));
    }

    __shared__ float s_sum[TPB];
    __shared__ float s_sq[TPB];
    __shared__ float s_mx[TPB / 32];
    __shared__ float s_mn[TPB / 32];

    s_sum[tid] = psum;
    s_sq[tid]  = psq;
    if (lane == 0) { s_mx[wid] = pmax; s_mn[wid] = pmin; }
    __syncthreads();

    // --- WMMA reduction of the 256 sum / 256 sumsq partials -----------------
    // D = A(16x4) * ones(4x16) + C : every column of D is the row-sum of A.
    // Slot assignment of the 64 LDS values into A is arbitrary (commutative).
    // All 8 waves execute identically -> EXEC is all-1s as WMMA requires.
    v2f ones; ones.x = 1.0f; ones.y = 1.0f;
    v8f accS = {};
    v8f accQ = {};
#pragma unroll
    for (int t = 0; t < 4; ++t) {
        const int base = t * 64;
        v2f aS, aQ;
        aS.x = s_sum[base + lane];      aS.y = s_sum[base + 32 + lane];
        aQ.x = s_sq[base + lane];       aQ.y = s_sq[base + 32 + lane];
        accS = __builtin_amdgcn_wmma_f32_16x16x4_f32(
            false, aS, false, ones, (short)0, accS, false, false);
        accQ = __builtin_amdgcn_wmma_f32_16x16x4_f32(
            false, aQ, false, ones, (short)0, accQ, false, false);
    }
    // lanes 0-15 hold D[M=0..7][N=lane] in VGPRs 0..7; lanes 16-31 hold M=8..15.
    float totS = ((accS[0] + accS[1]) + (accS[2] + accS[3])) +
                 ((accS[4] + accS[5]) + (accS[6] + accS[7]));
    float totQ = ((accQ[0] + accQ[1]) + (accQ[2] + accQ[3])) +
                 ((accQ[4] + accQ[5]) + (accQ[6] + accQ[7]));
    totS += __shfl_xor(totS, 16, 32);   // combine M=0..7 with M=8..15
    totQ += __shfl_xor(totQ, 16, 32);

    if (tid == 0) {
        float mx = s_mx[0], mn = s_mn[0];
#pragma unroll
        for (int i = 1; i < TPB / 32; ++i) {
            mx = fmaxf(mx, s_mx[i]);
            mn = fminf(mn, s_mn[i]);
        }
        const float N    = (float)HW;
        const float mean = totS / N;
        const float var  = (totQ - totS * totS / N) / (N - 1.0f);
        const float sd   = sqrtf(fmaxf(var, 0.0f));

        const int b = (int)blockIdx.x / NCH_C;
        const int c = (int)blockIdx.x % NCH_C;
        float* o = out + b * (4 * NCH_C);
        o[c]                = mean;
        o[NCH_C + c]        = sd;
        o[2 * NCH_C + c]    = mx;
        o[3 * NCH_C + c]    = mn;
    }
}

extern "C" void kernel_launch(void* const* d_in, const int* in_sizes, int n_in,
                              void* d_out, int out_size, void* d_ws, size_t ws_size,
                              hipStream_t stream) {
    (void)n_in; (void)d_ws; (void)ws_size; (void)out_size;
    const float* x   = (const float*)d_in[0];
    float*       out = (float*)d_out;
    const int nchan  = in_sizes[0] / HW;   // 32 * 16 = 512 workgroups
    image_stats_kernel<<<dim3(nchan), dim3(TPB), 0, stream>>>(x, out);
}